// MRR_10677288698543
// MI455X (gfx1250) — compile-verified
//
#include <hip/hip_runtime.h>
#include <hip/hip_bf16.h>

// ---------------------------------------------------------------------------
// MRR metric kernel for MI455X (gfx1250).
//
// pv[index[i]] = predict_val[i]  (index is a permutation)
// pos  = pv[0:P],  neg = pv[P:P+P*N] reshaped (P, N)
// rank[p] = 1 + sum_j (neg[p,j] > pos[p]);  out = (mean(1/rank), 1/rank)
//
// Memory-bound: ~262 MB mandatory HBM traffic -> ~11 us floor at 23.3 TB/s.
//   pass 1 : read ONLY index (RT hints -> stays resident in 192MB L2).
//   pass 1b: gather the P positive values (32 KB, cache-resident).
//   pass 2 : stream predict_val once (non-temporal) + re-read index (L2 hits);
//            p = (k-P)/num via host-precomputed magic multiply (3 VALU ops
//            instead of ~15 for a runtime u32 divide); counts via
//            non-returning u32 atomics (order-independent => deterministic).
//   pass 3 : one block: sample_mrr = 1/(1+count); deterministic fixed-order
//            mean; per-wave 32-lane sum via v_wmma_f32_16x16x4_f32 with an
//            all-ones B matrix (D[m][n] = partial[m] + partial[m+16]).
// ---------------------------------------------------------------------------

typedef float v2f __attribute__((ext_vector_type(2)));
typedef float v8f __attribute__((ext_vector_type(8)));
typedef float f4v __attribute__((ext_vector_type(4)));
typedef int   i4v __attribute__((ext_vector_type(4)));

__global__ void mrr_zero_counts(int* __restrict__ counts, int n) {
  int t = blockIdx.x * blockDim.x + threadIdx.x;
  if (t < n) counts[t] = 0;
}

// Pass 1: scan index only; remember source position of each positive sample.
__global__ void mrr_scan_index(const int* __restrict__ index,
                               int* __restrict__ pos_src,
                               int n_tot, int num_pos) {
  int t = blockIdx.x * blockDim.x + threadIdx.x;
  int i = t * 8;
  if (i + 7 < n_tot) {
    // Regular-temporal 128b loads: keep index hot in 192MB L2 for pass 2.
    i4v k0 = *(const i4v*)(index + i);
    i4v k1 = *(const i4v*)(index + i + 4);
    if (k0.x < num_pos) pos_src[k0.x] = i + 0;
    if (k0.y < num_pos) pos_src[k0.y] = i + 1;
    if (k0.z < num_pos) pos_src[k0.z] = i + 2;
    if (k0.w < num_pos) pos_src[k0.w] = i + 3;
    if (k1.x < num_pos) pos_src[k1.x] = i + 4;
    if (k1.y < num_pos) pos_src[k1.y] = i + 5;
    if (k1.z < num_pos) pos_src[k1.z] = i + 6;
    if (k1.w < num_pos) pos_src[k1.w] = i + 7;
  } else {
    for (; i < n_tot; ++i) {
      int k = index[i];
      if (k < num_pos) pos_src[k] = i;
    }
  }
}

// Pass 1b: gather positive values into a dense, cache-resident table.
__global__ void mrr_gather_pos(const float* __restrict__ pv,
                               const int* __restrict__ pos_src,
                               float* __restrict__ pos_val, int num_pos) {
  int p = blockIdx.x * blockDim.x + threadIdx.x;
  if (p < num_pos) pos_val[p] = pv[pos_src[p]];
}

__device__ __forceinline__ void mrr_one(int k, float v, int num_pos, unsigned num,
                                        unsigned long long magic, int shift,
                                        const float* __restrict__ pos_val,
                                        int* __restrict__ counts) {
  if (k >= num_pos) {
    unsigned x = (unsigned)(k - num_pos);
    unsigned p;
    if (magic) {
      // q = floor(x / num) via host-verified magic: one mad_u64_u32 + shift.
      p = (unsigned)(((unsigned long long)x * magic) >> shift);
      if ((unsigned long long)p * num > x) --p;   // 2-op safety belt
    } else {
      p = x / num;                                // fallback (host found no magic)
    }
    if (v > pos_val[p]) atomicAdd(&counts[p], 1); // non-returning u32 atomic
  }
}

// Pass 2: stream both arrays; count negatives beating their positive.
__global__ void mrr_count(const float* __restrict__ pv,
                          const int* __restrict__ index,
                          const float* __restrict__ pos_val,
                          int* __restrict__ counts,
                          int n_tot, int num_pos, unsigned num,
                          unsigned long long magic, int shift) {
  int t = blockIdx.x * blockDim.x + threadIdx.x;
  int i = t * 8;
  if (i + 7 < n_tot) {
    i4v k0 = *(const i4v*)(index + i);                            // L2 hits
    i4v k1 = *(const i4v*)(index + i + 4);
    f4v v0 = __builtin_nontemporal_load((const f4v*)(pv + i));    // NT stream:
    f4v v1 = __builtin_nontemporal_load((const f4v*)(pv + i + 4));// read once,
                                                                  // keep index in L2
    mrr_one(k0.x, v0.x, num_pos, num, magic, shift, pos_val, counts);
    mrr_one(k0.y, v0.y, num_pos, num, magic, shift, pos_val, counts);
    mrr_one(k0.z, v0.z, num_pos, num, magic, shift, pos_val, counts);
    mrr_one(k0.w, v0.w, num_pos, num, magic, shift, pos_val, counts);
    mrr_one(k1.x, v1.x, num_pos, num, magic, shift, pos_val, counts);
    mrr_one(k1.y, v1.y, num_pos, num, magic, shift, pos_val, counts);
    mrr_one(k1.z, v1.z, num_pos, num, magic, shift, pos_val, counts);
    mrr_one(k1.w, v1.w, num_pos, num, magic, shift, pos_val, counts);
  } else {
    for (; i < n_tot; ++i)
      mrr_one(index[i], pv[i], num_pos, num, magic, shift, pos_val, counts);
  }
}

// Pass 3: sample_mrr + deterministic mean. One block, 256 threads (8 wave32).
// Per-wave cross-lane sum via V_WMMA_F32_16X16X4_F32 with B = ones:
// lanes 0-15 supply A[m][K0]=partial(m), lanes 16-31 supply A[m][K2]=
// partial(m+16); with B all-ones D[m][n] = partial[m]+partial[m+16], so
// sum_j c[j] gives the half-sums and one shfl_xor(16) completes the wave sum.
__global__ void mrr_finalize(const int* __restrict__ counts,
                             float* __restrict__ out, int num_pos) {
  __shared__ float wsum[8];
  const int t = threadIdx.x;
  float local = 0.0f;
  for (int idx = t; idx < num_pos; idx += 256) {
    float r = 1.0f / (float)(1 + counts[idx]);
    out[1 + idx] = r;                  // sample_mrr
    local += r;
  }

  v2f a; a.x = local; a.y = 0.0f;      // A: lane partials
  v2f b; b.x = 1.0f;  b.y = 1.0f;      // B: all ones
  v8f c = {};
  c = __builtin_amdgcn_wmma_f32_16x16x4_f32(
      false, a, false, b, (short)0, c, false, false);

  float s = c[0] + c[1] + c[2] + c[3] + c[4] + c[5] + c[6] + c[7];
  s += __shfl_xor(s, 16, 32);          // combine row halves -> full wave sum

  const int wave = t >> 5, lane = t & 31;
  if (lane == 0) wsum[wave] = s;
  __syncthreads();
  if (t == 0) {
    float tot = 0.0f;
    for (int w = 0; w < (int)(blockDim.x >> 5); ++w) tot += wsum[w];
    out[0] = tot / (float)num_pos;     // mrr
  }
}

extern "C" void kernel_launch(void* const* d_in, const int* in_sizes, int n_in,
                              void* d_out, int out_size, void* d_ws, size_t ws_size,
                              hipStream_t stream) {
  const float* pv    = (const float*)d_in[0];
  const int*   index = (const int*)d_in[1];   // integer input -> const int*

  const int n_tot   = in_sizes[0];
  const int num_pos = out_size - 1;                             // 8192
  const unsigned num = (unsigned)((n_tot - num_pos) / num_pos); // 4000

  // Host-side magic for exact floor(x/num), x in [0, x_max]:
  // M = floor(2^S/num)+1, residual r = M*num - 2^S in (0, num].
  // Exact for all x <= x_max iff x_max * r < 2^S (then the +delta never
  // crosses a multiple of num). Also require M < 2^32 so x*M fits in 64 bits.
  unsigned long long magic = 0; int shift = 0;
  {
    const unsigned long long x_max = (unsigned long long)n_tot;
    for (int S = 33; S <= 62 && num > 1; ++S) {
      unsigned long long pow2 = 1ULL << S;
      unsigned long long M = pow2 / num + 1ULL;
      if (M >> 32) break;                        // product would overflow u64
      unsigned long long r = M * num - pow2;
      if (x_max * r < pow2) { magic = M; shift = S; break; }
    }
    if (num == 1) { magic = 1; shift = 0; }      // q = x
  }

  // Workspace layout (tiny: ~96 KB for num_pos=8192)
  int*   counts  = (int*)d_ws;
  int*   pos_src = counts + num_pos;
  float* pos_val = (float*)(pos_src + num_pos);

  const int tpb = 256;
  const int zb  = (num_pos + tpb - 1) / tpb;
  const int nvec = (n_tot + 7) / 8;
  const int gb  = (nvec + tpb - 1) / tpb;

  mrr_zero_counts<<<zb, tpb, 0, stream>>>(counts, num_pos);
  mrr_scan_index <<<gb, tpb, 0, stream>>>(index, pos_src, n_tot, num_pos);
  mrr_gather_pos <<<zb, tpb, 0, stream>>>(pv, pos_src, pos_val, num_pos);
  mrr_count      <<<gb, tpb, 0, stream>>>(pv, index, pos_val, counts,
                                          n_tot, num_pos, num, magic, shift);
  mrr_finalize   <<<1, 256, 0, stream>>>(counts, (float*)d_out, num_pos);
}